// MultiHeadAttentionLayer_6820408066290
// MI455X (gfx1250) — compile-verified
//
#include <hip/hip_runtime.h>

// ---------------------------------------------------------------------------
// MultiHeadAttention forward for MI455X (gfx1250), fp32 end-to-end.
//
// Roofline: ~86 GFLOP fp32 vs ~3 GB of HBM traffic (prob tensor is 1 GB and
// must be materialized) -> bandwidth-bound at ~130us @ 23.3 TB/s. fp32 WMMA
// (V_WMMA_F32_16X16X4_F32) is therefore the right precision: exact vs the
// fp32 reference, and compute is nowhere near the limiter.
// ---------------------------------------------------------------------------

typedef float v2f __attribute__((ext_vector_type(2)));
typedef float v8f __attribute__((ext_vector_type(8)));

#define D_MODEL 1024
#define HEADS   16
#define DK      64
#define T_SEQ   2048
#define BATCH   4
#define ROWS    (BATCH * T_SEQ)     // 8192
#define C3      (3 * D_MODEL)       // 3072

// D = A(16x4 f32) * B(4x16 f32) + C(16x16 f32), one matrix per wave32.
// Arg pattern mirrors probe-confirmed f16 variant:
// (neg_a, A, neg_b, B, c_mod, C, reuse_a, reuse_b)
__device__ __forceinline__ v8f wmma_f32(v2f a, v2f b, v8f c) {
  return __builtin_amdgcn_wmma_f32_16x16x4_f32(
      false, a, false, b, (short)0, c, false, false);
}

// ---------------------------------------------------------------------------
// Kernel 1: qkv = x @ W_qkv + b_qkv, scattered to head-major Q/K/V [B,H,T,dk].
// One wave computes one 16x16 tile of the [8192 x 3072] result.
// A 16-col tile never straddles a 64-col head boundary, so the whole tile
// lands in one segment/head.
// ---------------------------------------------------------------------------
__global__ __launch_bounds__(256) void qkv_gemm(
    const float* __restrict__ x, const float* __restrict__ W,
    const float* __restrict__ bias,
    float* __restrict__ Q, float* __restrict__ K, float* __restrict__ V) {
  const int lane = threadIdx.x & 31;
  const int wid  = blockIdx.x * (blockDim.x >> 5) + (threadIdx.x >> 5);
  const int TILES_N = C3 / 16;                 // 192
  const int TILES_M = ROWS / 16;               // 512
  if (wid >= TILES_M * TILES_N) return;        // wave-uniform
  const int tm = wid / TILES_N, tn = wid % TILES_N;
  const int half = lane >> 4, l16 = lane & 15;
  const int row0 = tm * 16, col0 = tn * 16;

  // A: lane l16 holds row m=l16; K offset 0,1 (lanes 0-15) or 2,3 (16-31).
  const float* arow = x + (size_t)(row0 + l16) * D_MODEL + 2 * half;
  // B: lane l16 holds col n=l16; rows kk+2*half, kk+2*half+1.
  const float* bcol = W + (size_t)(2 * half) * C3 + (col0 + l16);

  v8f acc = {};
  for (int kk = 0; kk < D_MODEL; kk += 4) {
    __builtin_prefetch(arow + kk + 64, 0, 1);  // global_prefetch_b8
    v2f a = *(const v2f*)(arow + kk);
    v2f b;
    b.x = bcol[(size_t)kk * C3];
    b.y = bcol[(size_t)(kk + 1) * C3];
    acc = wmma_f32(a, b, acc);
  }

  const float bv = bias[col0 + l16];
  const int seg = col0 / D_MODEL;              // 0=q 1=k 2=v
  const int hh  = (col0 % D_MODEL) / DK;
  const int dkc = (col0 & 63) + l16;           // column within head
  float* dst = (seg == 0) ? Q : (seg == 1) ? K : V;
#pragma unroll
  for (int r = 0; r < 8; ++r) {
    const int m = r + 8 * half;                // C/D layout: M = r + 8*half
    const int grow = row0 + m;
    const int b_ = grow >> 11;                 // / 2048
    const int t  = grow & 2047;
    dst[(((size_t)(b_ * HEADS + hh) * T_SEQ + t) * DK) + dkc] = acc[r] + bv;
  }
}

// ---------------------------------------------------------------------------
// Kernel 2: raw scores = (Q @ K^T) / 8, causal, into prob buffer.
// One wave per 16x16 tile; tiles fully above the diagonal are skipped
// (softmax kernel writes their zeros), diagonal tiles mask in-register.
// ---------------------------------------------------------------------------
__global__ __launch_bounds__(256) void attn_scores(
    const float* __restrict__ Q, const float* __restrict__ K,
    float* __restrict__ prob) {
  const int lane = threadIdx.x & 31;
  const int wid  = blockIdx.x * (blockDim.x >> 5) + (threadIdx.x >> 5);
  const int TT = T_SEQ / 16;                   // 128
  const int bh = wid / (TT * TT);
  if (bh >= BATCH * HEADS) return;
  const int rem = wid % (TT * TT);
  const int tq = rem / TT, tk = rem % TT;
  if (tk > tq) return;                         // fully masked tile (uniform)

  const int half = lane >> 4, l16 = lane & 15;
  const float* qrow = Q + (size_t)bh * T_SEQ * DK
                        + (size_t)(tq * 16 + l16) * DK + 2 * half;
  const float* krow = K + (size_t)bh * T_SEQ * DK
                        + (size_t)(tk * 16 + l16) * DK + 2 * half;
  v8f acc = {};
#pragma unroll
  for (int kk = 0; kk < DK; kk += 4) {
    v2f a = *(const v2f*)(qrow + kk);          // A = Q tile rows
    v2f b = *(const v2f*)(krow + kk);          // B = K^T -> K tile rows
    acc = wmma_f32(a, b, acc);
  }

  float* pb = prob + (size_t)bh * T_SEQ * T_SEQ;
  const int q0 = tq * 16, k0 = tk * 16;
#pragma unroll
  for (int r = 0; r < 8; ++r) {
    const int m = r + 8 * half;
    const int qg = q0 + m, kg = k0 + l16;
    float s = acc[r] * 0.125f;                 // 1/sqrt(64)
    if (kg > qg) s = -__builtin_inff();        // diagonal-tile mask
    pb[(size_t)qg * T_SEQ + kg] = s;
  }
}

// ---------------------------------------------------------------------------
// Kernel 3: in-place causal softmax, one 256-thread block per row.
// Reads only k <= q (valid region), writes zeros for k > q.
// ---------------------------------------------------------------------------
__global__ __launch_bounds__(256) void softmax_rows(float* __restrict__ prob) {
  __shared__ float red[256];
  const int row = blockIdx.x;                  // bh*T + q
  const int q   = row & (T_SEQ - 1);
  float* p = prob + (size_t)row * T_SEQ;
  const int tid = threadIdx.x;
  const int n = q + 1;                         // valid length

  float mx = -__builtin_inff();
  for (int i = tid; i < n; i += 256) mx = fmaxf(mx, p[i]);
  red[tid] = mx;
  __syncthreads();
  for (int s = 128; s > 0; s >>= 1) {
    if (tid < s) red[tid] = fmaxf(red[tid], red[tid + s]);
    __syncthreads();
  }
  mx = red[0];
  __syncthreads();

  float sum = 0.f;
  for (int i = tid; i < n; i += 256) sum += __expf(p[i] - mx);
  red[tid] = sum;
  __syncthreads();
  for (int s = 128; s > 0; s >>= 1) {
    if (tid < s) red[tid] += red[tid + s];
    __syncthreads();
  }
  const float inv = 1.0f / red[0];

  for (int i = tid; i < n; i += 256) p[i] = __expf(p[i] - mx) * inv;
  for (int i = n + tid; i < T_SEQ; i += 256) p[i] = 0.0f;
}

// ---------------------------------------------------------------------------
// Kernel 4: out = prob @ V, one wave per 16x16 tile of [T x dk] per (b,h).
// Causality: keys beyond q0+15 contribute zero prob, so truncate the K-loop.
// ---------------------------------------------------------------------------
__global__ __launch_bounds__(256) void attn_out(
    const float* __restrict__ prob, const float* __restrict__ V,
    float* __restrict__ out) {
  const int lane = threadIdx.x & 31;
  const int wid  = blockIdx.x * (blockDim.x >> 5) + (threadIdx.x >> 5);
  const int TT = T_SEQ / 16;                   // 128
  const int DT = DK / 16;                      // 4
  if (wid >= BATCH * HEADS * TT * DT) return;
  const int bh  = wid / (TT * DT);
  const int rem = wid % (TT * DT);
  const int tq = rem / DT, td = rem % DT;
  const int half = lane >> 4, l16 = lane & 15;

  const float* prow = prob + (size_t)bh * T_SEQ * T_SEQ
                           + (size_t)(tq * 16 + l16) * T_SEQ + 2 * half;
  const float* vcol = V + (size_t)bh * T_SEQ * DK
                        + (size_t)(2 * half) * DK + (td * 16 + l16);
  v8f acc = {};
  const int kend = tq * 16 + 16;               // causal truncation
  for (int kk = 0; kk < kend; kk += 4) {
    __builtin_prefetch(prow + kk + 64, 0, 1);
    v2f a = *(const v2f*)(prow + kk);
    v2f b;
    b.x = vcol[(size_t)kk * DK];
    b.y = vcol[(size_t)(kk + 1) * DK];
    acc = wmma_f32(a, b, acc);
  }

  const int b_ = bh / HEADS, h = bh % HEADS;
  const int q0 = tq * 16;
#pragma unroll
  for (int r = 0; r < 8; ++r) {
    const int m = r + 8 * half;
    out[((size_t)(b_ * T_SEQ + q0 + m) * D_MODEL) + h * DK + td * 16 + l16] =
        acc[r];
  }
}

// ---------------------------------------------------------------------------
extern "C" void kernel_launch(void* const* d_in, const int* in_sizes, int n_in,
                              void* d_out, int out_size, void* d_ws,
                              size_t ws_size, hipStream_t stream) {
  (void)in_sizes; (void)n_in; (void)out_size; (void)ws_size;
  const float* x     = (const float*)d_in[0];
  const float* W_qkv = (const float*)d_in[1];
  const float* b_qkv = (const float*)d_in[2];
  // d_in[3] = W_o, d_in[4] = b_o: reference never applies the output proj.

  float* out  = (float*)d_out;                          // [B,T,C]
  float* prob = out + (size_t)BATCH * T_SEQ * D_MODEL;  // [B,H,T,T]

  float* Q = (float*)d_ws;                              // [B,H,T,dk] each
  float* K = Q + (size_t)BATCH * HEADS * T_SEQ * DK;
  float* V = K + (size_t)BATCH * HEADS * T_SEQ * DK;    // total 96 MB

  // 1) QKV projection: 512*192 = 98304 tiles, 8 waves/block.
  qkv_gemm<<<dim3(98304 / 8), dim3(256), 0, stream>>>(x, W_qkv, b_qkv, Q, K, V);

  // 2) Causal scores: 64 * 128*128 waves (upper-triangle waves exit early).
  attn_scores<<<dim3(64 * 128 * 128 / 8), dim3(256), 0, stream>>>(Q, K, prob);

  // 3) Row softmax: one block per (b,h,q) row.
  softmax_rows<<<dim3(BATCH * HEADS * T_SEQ), dim3(256), 0, stream>>>(prob);

  // 4) prob @ V: 64 * 128 * 4 = 32768 tiles.
  attn_out<<<dim3(32768 / 8), dim3(256), 0, stream>>>(prob, V, out);
}